// QCNN_13924283973958
// MI455X (gfx1250) — compile-verified
//
#include <hip/hip_runtime.h>

typedef __attribute__((ext_vector_type(2))) float v2f;
typedef __attribute__((ext_vector_type(8))) float v8f;

#define NQ 13
#define DIM 8192          // 2^13
#define BATCH 2
#define RHO_STRIDE 16384  // 128*128 per-batch stride for rho buffers

// ---- workspace layout (float offsets) ----
// 0      : U_re[4][16]
// 64     : U_im[4][16]
// 128    : psi_re[2][8192]
// 16512  : psi_im[2][8192]
// 32896  : rhoA_re[2][16384]
// 65664  : rhoA_im[2][16384]
// 98432  : rhoB_re[2][16384]
// 131200 : rhoB_im[2][16384]   (end = 163968 floats ~ 656 KB)
#define WS_U_RE   0
#define WS_U_IM   64
#define WS_PSI_RE 128
#define WS_PSI_IM 16512
#define WS_RA_RE  32896
#define WS_RA_IM  65664
#define WS_RB_RE  98432
#define WS_RB_IM  131200

// ---------------- expm of 4x4 anti-Hermitian (scaling & squaring Taylor) ----
__global__ void k_expm(const float* Hre, const float* Him, float* ws) {
  int layer = threadIdx.x;
  if (layer >= 4) return;
  double Are[4][4], Aim[4][4];
  for (int i = 0; i < 4; i++)
    for (int j = 0; j < 4; j++) {
      Are[i][j] = (double)Hre[layer*16 + i*4 + j] - (double)Hre[layer*16 + j*4 + i];
      Aim[i][j] = (double)Him[layer*16 + i*4 + j] + (double)Him[layer*16 + j*4 + i];
    }
  double m = 0.0;
  for (int i = 0; i < 4; i++)
    for (int j = 0; j < 4; j++) {
      double v = fabs(Are[i][j]) + fabs(Aim[i][j]);
      if (v > m) m = v;
    }
  int s = 0;
  while (m > 0.5 && s < 40) { m *= 0.5; s++; }
  double sc = 1.0;
  for (int k = 0; k < s; k++) sc *= 0.5;
  for (int i = 0; i < 4; i++)
    for (int j = 0; j < 4; j++) { Are[i][j] *= sc; Aim[i][j] *= sc; }
  double Ere[4][4], Eim[4][4], Tre[4][4], Tim[4][4];
  for (int i = 0; i < 4; i++)
    for (int j = 0; j < 4; j++) {
      Ere[i][j] = (i == j) ? 1.0 : 0.0; Eim[i][j] = 0.0;
      Tre[i][j] = Ere[i][j];            Tim[i][j] = 0.0;
    }
  for (int k = 1; k <= 20; k++) {
    double Nre[4][4], Nim[4][4];
    for (int i = 0; i < 4; i++)
      for (int j = 0; j < 4; j++) {
        double sr = 0, si = 0;
        for (int c = 0; c < 4; c++) {
          sr += Tre[i][c]*Are[c][j] - Tim[i][c]*Aim[c][j];
          si += Tre[i][c]*Aim[c][j] + Tim[i][c]*Are[c][j];
        }
        Nre[i][j] = sr / k; Nim[i][j] = si / k;
      }
    for (int i = 0; i < 4; i++)
      for (int j = 0; j < 4; j++) {
        Tre[i][j] = Nre[i][j]; Tim[i][j] = Nim[i][j];
        Ere[i][j] += Nre[i][j]; Eim[i][j] += Nim[i][j];
      }
  }
  for (int q = 0; q < s; q++) {
    double Nre[4][4], Nim[4][4];
    for (int i = 0; i < 4; i++)
      for (int j = 0; j < 4; j++) {
        double sr = 0, si = 0;
        for (int c = 0; c < 4; c++) {
          sr += Ere[i][c]*Ere[c][j] - Eim[i][c]*Eim[c][j];
          si += Ere[i][c]*Eim[c][j] + Eim[i][c]*Ere[c][j];
        }
        Nre[i][j] = sr; Nim[i][j] = si;
      }
    for (int i = 0; i < 4; i++)
      for (int j = 0; j < 4; j++) { Ere[i][j] = Nre[i][j]; Eim[i][j] = Nim[i][j]; }
  }
  float* Ure = ws + WS_U_RE + layer*16;
  float* Uim = ws + WS_U_IM + layer*16;
  for (int i = 0; i < 4; i++)
    for (int j = 0; j < 4; j++) {
      Ure[i*4 + j] = (float)Ere[i][j];
      Uim[i*4 + j] = (float)Eim[i][j];
    }
}

// ---------------- normalize psi into workspace ----------------
__global__ void k_normalize(const float* pre, const float* pim, float* ws) {
  int b = blockIdx.x;
  __shared__ float red[256];
  const float* r = pre + b*DIM;
  const float* im = pim + b*DIM;
  float s = 0.f;
  for (int k = threadIdx.x; k < DIM; k += 256) s += r[k]*r[k] + im[k]*im[k];
  red[threadIdx.x] = s;
  __syncthreads();
  for (int off = 128; off > 0; off >>= 1) {
    if (threadIdx.x < off) red[threadIdx.x] += red[threadIdx.x + off];
    __syncthreads();
  }
  float inv = rsqrtf(red[0]);
  float* wr = ws + WS_PSI_RE + b*DIM;
  float* wi = ws + WS_PSI_IM + b*DIM;
  for (int k = threadIdx.x; k < DIM; k += 256) { wr[k] = r[k]*inv; wi[k] = im[k]*inv; }
}

// insert two zero bits at positions lo<hi
__device__ __forceinline__ int expand2(int x, int lo, int hi) {
  int t = ((x >> lo) << (lo + 1)) | (x & ((1 << lo) - 1));
  t = ((t >> hi) << (hi + 1)) | (t & ((1 << hi) - 1));
  return t;
}

// ---------------- layer-0 two-qubit gate on pure state (in-place) ----------
__global__ void k_conv_state(float* ws, int layer, int q0, int q1) {
  int tid = blockIdx.x*blockDim.x + threadIdx.x;  // 2 batches * 2048 groups
  int b = tid >> 11;
  int g = tid & 2047;
  const float* Ure = ws + WS_U_RE + layer*16;
  const float* Uim = ws + WS_U_IM + layer*16;
  float* pr = ws + WS_PSI_RE + b*DIM;
  float* pi = ws + WS_PSI_IM + b*DIM;
  int p0 = 12 - q0, p1 = 12 - q1;
  int lo = p0 < p1 ? p0 : p1, hi = p0 < p1 ? p1 : p0;
  int base = expand2(g, lo, hi);
  int idx[4];
  for (int t = 0; t < 4; t++) idx[t] = base | ((t >> 1) << p0) | ((t & 1) << p1);
  float vr[4], vi[4];
  for (int t = 0; t < 4; t++) { vr[t] = pr[idx[t]]; vi[t] = pi[idx[t]]; }
  for (int o = 0; o < 4; o++) {
    float sr = 0, si = 0;
    for (int j = 0; j < 4; j++) {
      float ur = Ure[o*4 + j], ui = Uim[o*4 + j];
      sr += ur*vr[j] - ui*vi[j];
      si += ur*vi[j] + ui*vr[j];
    }
    pr[idx[o]] = sr; pi[idx[o]] = si;
  }
}

// psi flat bit layout: qubit k = bit (12-k).  a = kept (1,3,5,7,9,11,12),
// t = traced (0,2,4,6,8,10), both MSB-first in qubit order.
__device__ __forceinline__ int ileave(int t, int a) {
  return ((t >> 5 & 1) << 12) | ((a >> 6 & 1) << 11) | ((t >> 4 & 1) << 10) |
         ((a >> 5 & 1) << 9)  | ((t >> 3 & 1) << 8)  | ((a >> 4 & 1) << 7)  |
         ((t >> 2 & 1) << 6)  | ((a >> 3 & 1) << 5)  | ((t >> 1 & 1) << 4)  |
         ((a >> 2 & 1) << 3)  | ((t & 1) << 2)       | ((a >> 1 & 1) << 1)  | (a & 1);
}

// ---------------- pooled density matrix rho[a,b] = sum_t psi(t,a) psi(t,b)^*
// via V_WMMA_F32_16X16X4_F32: one wave per 16x16 tile, K=64 in 16 chunks of 4.
__global__ void k_gram(float* ws) {
  int blk = blockIdx.x;          // 2 batches * 64 tiles
  int b = blk >> 6;
  int tile = blk & 63;
  int at = tile >> 3, bt = tile & 7;
  int L = threadIdx.x;           // single wave32, EXEC all ones
  const float* pr = ws + WS_PSI_RE + b*DIM;
  const float* pi = ws + WS_PSI_IM + b*DIM;
  int row = L & 15;
  int k0 = (L >> 4) << 1;        // lanes 0-15: K=0,1 ; lanes 16-31: K=2,3
  int aIdx = at*16 + row;        // A fragment: lanes = M (rows of output)
  int bIdx = bt*16 + row;        // B fragment: lanes = N (cols of output)
  v8f accR = {0, 0, 0, 0, 0, 0, 0, 0};
  v8f accI = {0, 0, 0, 0, 0, 0, 0, 0};
  for (int c = 0; c < 16; c++) {
    int t0 = c*4 + k0, t1 = t0 + 1;
    int ia0 = ileave(t0, aIdx), ia1 = ileave(t1, aIdx);
    int ib0 = ileave(t0, bIdx), ib1 = ileave(t1, bIdx);
    v2f Ar = { pr[ia0], pr[ia1] };
    v2f Ai = { pi[ia0], pi[ia1] };
    v2f Br = { pr[ib0], pr[ib1] };
    v2f Bi = { pi[ib0], pi[ib1] };
    v2f nAr = { -Ar.x, -Ar.y };  // f32 WMMA has no A/B NEG modifier
    // rho_re += Are*Bre + Aim*Bim ; rho_im += Aim*Bre - Are*Bim
    accR = __builtin_amdgcn_wmma_f32_16x16x4_f32(false, Ar,  false, Br, (short)0, accR, false, false);
    accR = __builtin_amdgcn_wmma_f32_16x16x4_f32(false, Ai,  false, Bi, (short)0, accR, false, false);
    accI = __builtin_amdgcn_wmma_f32_16x16x4_f32(false, Ai,  false, Br, (short)0, accI, false, false);
    accI = __builtin_amdgcn_wmma_f32_16x16x4_f32(false, nAr, false, Bi, (short)0, accI, false, false);
  }
  float* rr = ws + WS_RA_RE + b*RHO_STRIDE;
  float* ri = ws + WS_RA_IM + b*RHO_STRIDE;
  int N = L & 15;
  int mbase = (L >> 4) ? 8 : 0;  // C/D layout: vgpr r -> M=r (lanes<16) / r+8
  for (int r = 0; r < 8; r++) {
    int a = at*16 + mbase + r;
    int bb = bt*16 + N;
    rr[a*128 + bb] = accR[r];
    ri[a*128 + bb] = accI[r];
  }
}

// ---------------- two-qubit gate on density matrix: Y = U X U^dagger -------
__global__ void k_conv_dm(float* rre, float* rim, const float* ws,
                          int layer, int n, int q0, int q1) {
  int nrest = n - 2;
  int total = BATCH << (2*nrest);
  int tid = blockIdx.x*blockDim.x + threadIdx.x;
  if (tid >= total) return;
  int b = tid >> (2*nrest);
  int rem = tid & ((1 << (2*nrest)) - 1);
  int rrest = rem >> nrest;
  int crest = rem & ((1 << nrest) - 1);
  const float* Ure = ws + WS_U_RE + layer*16;
  const float* Uim = ws + WS_U_IM + layer*16;
  int p0 = n - 1 - q0, p1 = n - 1 - q1;
  int lo = p0 < p1 ? p0 : p1, hi = p0 < p1 ? p1 : p0;
  int dim = 1 << n;
  int rb = expand2(rrest, lo, hi);
  int cb = expand2(crest, lo, hi);
  int ridx[4], cidx[4];
  for (int t = 0; t < 4; t++) {
    ridx[t] = rb | ((t >> 1) << p0) | ((t & 1) << p1);
    cidx[t] = cb | ((t >> 1) << p0) | ((t & 1) << p1);
  }
  float* Rr = rre + b*RHO_STRIDE;
  float* Ri = rim + b*RHO_STRIDE;
  float Xr[4][4], Xi[4][4];
  for (int i = 0; i < 4; i++)
    for (int j = 0; j < 4; j++) {
      Xr[i][j] = Rr[ridx[i]*dim + cidx[j]];
      Xi[i][j] = Ri[ridx[i]*dim + cidx[j]];
    }
  float Zr[4][4], Zi[4][4];
  for (int i = 0; i < 4; i++)
    for (int j = 0; j < 4; j++) {
      float sr = 0, si = 0;
      for (int c = 0; c < 4; c++) {
        float ur = Ure[i*4 + c], ui = Uim[i*4 + c];
        sr += ur*Xr[c][j] - ui*Xi[c][j];
        si += ur*Xi[c][j] + ui*Xr[c][j];
      }
      Zr[i][j] = sr; Zi[i][j] = si;
    }
  for (int i = 0; i < 4; i++)
    for (int k = 0; k < 4; k++) {
      float sr = 0, si = 0;
      for (int c = 0; c < 4; c++) {
        float ur = Ure[k*4 + c], ui = Uim[k*4 + c];  // conj(U[k][c])
        sr += Zr[i][c]*ur + Zi[i][c]*ui;
        si += Zi[i][c]*ur - Zr[i][c]*ui;
      }
      Rr[ridx[i]*dim + cidx[k]] = sr;
      Ri[ridx[i]*dim + cidx[k]] = si;
    }
}

__device__ __forceinline__ int pdep_sw(int v, int mask) {
  int r = 0, b = 0;
  for (int i = 0; i < 13; i++)
    if (mask & (1 << i)) { r |= ((v >> b) & 1) << i; b++; }
  return r;
}

// ---------------- partial trace (pool): trace row qubit q with col qubit q -
__global__ void k_pool(const float* sre, const float* sim, float* dre, float* dim_,
                       int n_old, int n_new, int tracedMask, int keptMask) {
  int nn = 1 << n_new;
  int total = BATCH*nn*nn;
  int tid = blockIdx.x*blockDim.x + threadIdx.x;
  if (tid >= total) return;
  int b = tid / (nn*nn);
  int rem = tid - b*nn*nn;
  int a = rem / nn, bb = rem % nn;
  int dimo = 1 << n_old;
  int ra = pdep_sw(a, keptMask);
  int cb = pdep_sw(bb, keptMask);
  int nt = 1 << (n_old - n_new);
  const float* Sr = sre + b*RHO_STRIDE;
  const float* Si = sim + b*RHO_STRIDE;
  float sr = 0, si = 0;
  for (int t = 0; t < nt; t++) {
    int tt = pdep_sw(t, tracedMask);
    sr += Sr[(ra | tt)*dimo + (cb | tt)];
    si += Si[(ra | tt)*dimo + (cb | tt)];
  }
  dre[b*RHO_STRIDE + a*nn + bb] = sr;
  dim_[b*RHO_STRIDE + a*nn + bb] = si;
}

__global__ void k_final(const float* rre, float* out) {
  int b = threadIdx.x;
  if (b < BATCH) out[b] = rre[b*RHO_STRIDE];  // rho[0,0].real
}

extern "C" void kernel_launch(void* const* d_in, const int* in_sizes, int n_in,
                              void* d_out, int out_size, void* d_ws, size_t ws_size,
                              hipStream_t stream) {
  const float* psi_re = (const float*)d_in[0];
  const float* psi_im = (const float*)d_in[1];
  const float* H_re   = (const float*)d_in[2];
  const float* H_im   = (const float*)d_in[3];
  float* ws  = (float*)d_ws;
  float* out = (float*)d_out;

  float* rhoA_re = ws + WS_RA_RE; float* rhoA_im = ws + WS_RA_IM;
  float* rhoB_re = ws + WS_RB_RE; float* rhoB_im = ws + WS_RB_IM;

  k_expm<<<1, 4, 0, stream>>>(H_re, H_im, ws);
  k_normalize<<<2, 256, 0, stream>>>(psi_re, psi_im, ws);

  static const int pairs0[12][2] = {{0,1},{2,3},{4,5},{6,7},{8,9},{10,11},
                                    {1,2},{3,4},{5,6},{7,8},{9,10},{11,12}};
  for (int g = 0; g < 12; ++g)
    k_conv_state<<<16, 256, 0, stream>>>(ws, 0, pairs0[g][0], pairs0[g][1]);

  // pooled rho (128x128 per batch) directly from psi via f32 WMMA Gram
  k_gram<<<128, 32, 0, stream>>>(ws);

  static const int pairs1[6][2] = {{0,1},{2,3},{4,5},{1,2},{3,4},{5,6}};
  for (int g = 0; g < 6; ++g)
    k_conv_dm<<<8, 256, 0, stream>>>(rhoA_re, rhoA_im, ws, 1, 7, pairs1[g][0], pairs1[g][1]);
  // n=7 -> 4 : trace qubits {0,2,4} = bits {6,4,2}
  k_pool<<<2, 256, 0, stream>>>(rhoA_re, rhoA_im, rhoB_re, rhoB_im, 7, 4, 0x54, 0x2B);

  static const int pairs2[4][2] = {{0,1},{2,3},{1,2},{3,0}};
  for (int g = 0; g < 4; ++g)
    k_conv_dm<<<1, 32, 0, stream>>>(rhoB_re, rhoB_im, ws, 2, 4, pairs2[g][0], pairs2[g][1]);
  // n=4 -> 2 : trace qubits {0,2} = bits {3,1}
  k_pool<<<1, 32, 0, stream>>>(rhoB_re, rhoB_im, rhoA_re, rhoA_im, 4, 2, 0xA, 0x5);

  k_conv_dm<<<1, 32, 0, stream>>>(rhoA_re, rhoA_im, ws, 3, 2, 0, 1);
  // n=2 -> 1 : trace qubit {0} = bit {1}
  k_pool<<<1, 8, 0, stream>>>(rhoA_re, rhoA_im, rhoB_re, rhoB_im, 2, 1, 0x2, 0x1);

  k_final<<<1, 2, 0, stream>>>(rhoB_re, out);
}